// ScaledDotProductAttention_46033459478807
// MI455X (gfx1250) — compile-verified
//
#include <hip/hip_runtime.h>

// MI455X / gfx1250, wave32. FP32 WMMA (V_WMMA_F32_16X16X4_F32) for both GEMMs
// to match the fp32 reference exactly; entmax-1.5 tau found by safeguarded
// Newton on f(tau) = sum(max(x - tau,0)^2) - 1 (convex, decreasing on [-1,0];
// Newton from tau=-1 converges monotonically + quadratically).

typedef __attribute__((ext_vector_type(2))) float v2f;
typedef __attribute__((ext_vector_type(8))) float v8f;

#define S_LEN 1024
#define D_DIM 64
#define BH    64   // B*H = 8*8

__global__ __launch_bounds__(256)
void entmax_attn_kernel(const float* __restrict__ q,
                        const float* __restrict__ k,
                        const float* __restrict__ v,
                        float* __restrict__ out)
{
    // 16 rows x 1024 cols of scores/probs; head reused for partial-C reduction.
    __shared__ float sm[16 * S_LEN];   // 64 KB

    const int lane  = threadIdx.x & 31;
    const int wave  = threadIdx.x >> 5;     // 8 waves
    const int lmod  = lane & 15;
    const int lhalf = lane >> 4;

    const int bh    = blockIdx.y;           // head index (B*H)
    const int qbase = blockIdx.x * 16;      // query tile base

    // ---------------- Phase 1: x = (Q K^T) * (1/sqrt(D)) * 0.5 -> LDS ------
    // f32 A-fragment layout (16x4): lane(0..15)=row M, VGPR0/1 = K{0,1};
    // lanes 16..31 same rows, K{2,3}.  K-offset = 4*s + 2*lhalf + {0,1}.
    const float* qp = q + ((size_t)(bh * S_LEN + qbase + lmod)) * D_DIM;
    v2f aq[16];
#pragma unroll
    for (int s = 0; s < 16; ++s)
        aq[s] = *(const v2f*)(qp + 4 * s + 2 * lhalf);

#pragma unroll
    for (int t = 0; t < 8; ++t) {
        const int nbase = (wave * 8 + t) * 16;           // key tile
        const float* kp = k + ((size_t)(bh * S_LEN + nbase + lmod)) * D_DIM;
        v8f acc = {};
#pragma unroll
        for (int s = 0; s < 16; ++s) {
            v2f bf = *(const v2f*)(kp + 4 * s + 2 * lhalf);
            acc = __builtin_amdgcn_wmma_f32_16x16x4_f32(
                false, aq[s], false, bf, (short)0, acc, false, false);
        }
        // C layout: VGPR r -> row r + 8*lhalf, col = lmod
        // fold score scale 1/8 and entmax's x/2:  * 0.0625
#pragma unroll
        for (int r = 0; r < 8; ++r)
            sm[(r + 8 * lhalf) * S_LEN + nbase + lmod] = acc[r] * 0.0625f;
    }
    __syncthreads();

    // ---------------- Phase 2: entmax-1.5 per row (Newton on tau) ----------
    // wave w owns rows 2w, 2w+1; 32 elements per lane in registers.
    for (int rr = 0; rr < 2; ++rr) {
        const int row = wave * 2 + rr;
        float* rp = &sm[row * S_LEN];
        float x[32];
#pragma unroll
        for (int j = 0; j < 32; ++j) x[j] = rp[lane + 32 * j];

        float m = x[0];
#pragma unroll
        for (int j = 1; j < 32; ++j) m = fmaxf(m, x[j]);
#pragma unroll
        for (int off = 16; off >= 1; off >>= 1)
            m = fmaxf(m, __shfl_xor(m, off, 32));
#pragma unroll
        for (int j = 0; j < 32; ++j) x[j] -= m;      // max(x) == 0 now

        // f(tau) = sum((x-tau)_+^2) - 1: convex, decreasing; root in [-1, 0).
        // f(-1) >= 0, so Newton from tau=-1 approaches the root from the left
        // monotonically with quadratic convergence.
        float tau = -1.0f;
        for (int it = 0; it < 12; ++it) {
            float s = 0.f, g = 0.f;
#pragma unroll
            for (int j = 0; j < 32; ++j) {
                float d = fmaxf(x[j] - tau, 0.f);
                s = fmaf(d, d, s);      // f  contribution
                g += d;                 // -f'/2 contribution
            }
#pragma unroll
            for (int off = 16; off >= 1; off >>= 1) {
                s += __shfl_xor(s, off, 32);
                g += __shfl_xor(g, off, 32);
            }
            g = fmaxf(g, 1e-20f);       // f' = -2g; g > 0 while tau < 0
            tau += (s - 1.0f) / (2.0f * g);   // tau - f/f'
        }

        float p[32];
        float ssum = 0.f;
#pragma unroll
        for (int j = 0; j < 32; ++j) {
            float d = fmaxf(x[j] - tau, 0.f);
            p[j] = d * d;
            ssum += p[j];
        }
#pragma unroll
        for (int off = 16; off >= 1; off >>= 1)
            ssum += __shfl_xor(ssum, off, 32);
        const float inv = (ssum > 0.f) ? (1.0f / ssum) : 0.f;
#pragma unroll
        for (int j = 0; j < 32; ++j) rp[lane + 32 * j] = p[j] * inv;
    }
    __syncthreads();

    // ---------------- Phase 3: out = P (16x1024) @ V (1024x64) -------------
    // 4 output tiles (N) x 2 K-halves mapped onto 8 waves.
    const int t      = wave & 3;
    const int kstart = (wave >> 2) * 512;
    const int ncol   = t * 16 + lmod;
    const float* vp  = v + ((size_t)bh * S_LEN) * D_DIM + ncol;

    v8f acc = {};
#pragma unroll 8
    for (int kb = kstart; kb < kstart + 512; kb += 4) {
        v2f a = *(const v2f*)(&sm[lmod * S_LEN + kb + 2 * lhalf]);
        v2f b;
        b.x = vp[(size_t)(kb + 2 * lhalf + 0) * D_DIM];
        b.y = vp[(size_t)(kb + 2 * lhalf + 1) * D_DIM];
        acc = __builtin_amdgcn_wmma_f32_16x16x4_f32(
            false, a, false, b, (short)0, acc, false, false);
    }
    __syncthreads();                 // all P reads done; safe to reuse LDS

#pragma unroll
    for (int r = 0; r < 8; ++r)
        sm[wave * 256 + r * 32 + lane] = acc[r];
    __syncthreads();

    if (wave < 4) {                  // combine the two K-halves, store
#pragma unroll
        for (int r = 0; r < 8; ++r) {
            float val = sm[wave * 256 + r * 32 + lane]
                      + sm[(wave + 4) * 256 + r * 32 + lane];
            const int row = r + 8 * lhalf;
            const int col = wave * 16 + lmod;
            out[((size_t)(bh * S_LEN + qbase + row)) * D_DIM + col] = val;
        }
    }
}

extern "C" void kernel_launch(void* const* d_in, const int* in_sizes, int n_in,
                              void* d_out, int out_size, void* d_ws, size_t ws_size,
                              hipStream_t stream) {
    (void)in_sizes; (void)n_in; (void)out_size; (void)d_ws; (void)ws_size;
    const float* q = (const float*)d_in[0];
    const float* k = (const float*)d_in[1];
    const float* v = (const float*)d_in[2];
    float* out = (float*)d_out;

    dim3 grid(S_LEN / 16, BH);   // 64 query tiles x 64 heads = 4096 blocks
    entmax_attn_kernel<<<grid, 256, 0, stream>>>(q, k, v, out);
}